// Attention_5729486373510
// MI455X (gfx1250) — compile-verified
//
#include <hip/hip_runtime.h>
#include <hip/hip_bf16.h>
#include <math.h>

typedef __attribute__((ext_vector_type(2))) float v2f;
typedef __attribute__((ext_vector_type(4))) float f4;
typedef __attribute__((ext_vector_type(8))) float v8f;

#define HW   16384   // 128*128
#define CH   64      // channel dim (q/k extent)
#define LDSP 68      // padded LDS row stride (floats); 272B row = 17*16B (b128-aligned)

// ---------------- zero the score workspace (16*64*64 floats) ----------------
__global__ void zero_scores_kernel(float* __restrict__ s) {
    int idx = blockIdx.x * 256 + threadIdx.x;   // grid 256x256 == 65536
    s[idx] = 0.0f;
}

// ---------------- phase 1: scores[b,q,k] += sum_hw Q[b,hw,q]*K[b,hw,k] ------
constexpr int P1_SPLITS = 64;                 // workgroups per batch
constexpr int P1_ROWS   = HW / P1_SPLITS;     // 256 hw rows per WG
constexpr int P1_CHUNK  = 32;                 // hw rows staged in LDS per iter

__global__ __launch_bounds__(512) void scores_kernel(const float* __restrict__ Q,
                                                     const float* __restrict__ K,
                                                     float* __restrict__ scores) {
    __shared__ float Qs[P1_CHUNK][LDSP];
    __shared__ float Ks[P1_CHUNK][LDSP];

    const int b     = blockIdx.x / P1_SPLITS;
    const int split = blockIdx.x % P1_SPLITS;
    const int tid   = threadIdx.x;
    const int wave  = tid >> 5;          // 0..15
    const int lane  = tid & 31;
    const int m     = lane & 15;         // A/B row-or-col index within tile
    const int half  = lane >> 4;         // k-group select (K = v + 2*half)
    const int q0    = (wave >> 2) * 16;  // 4x4 grid of 16x16 tiles
    const int k0    = (wave & 3) * 16;

    const float* Qb = Q + (size_t)b * HW * CH;
    const float* Kb = K + (size_t)b * HW * CH;

    // vectorized staging coords: 32 rows x 16 float4 == 512 threads
    const int sr = tid >> 4;             // staging row   0..31
    const int sc = (tid & 15) * 4;       // staging col   0,4,...,60

    v8f acc = {};
    const int hw_base = split * P1_ROWS;

    for (int c0 = 0; c0 < P1_ROWS; c0 += P1_CHUNK) {
        // one nontemporal b128 of Q and K per thread (streaming: read once)
        size_t g = (size_t)(hw_base + c0 + sr) * CH + sc;
        f4 qv = __builtin_nontemporal_load((const f4*)(Qb + g));
        f4 kv = __builtin_nontemporal_load((const f4*)(Kb + g));
        *(f4*)&Qs[sr][sc] = qv;
        *(f4*)&Ks[sr][sc] = kv;
        __syncthreads();

        #pragma unroll
        for (int kk = 0; kk < P1_CHUNK / 4; ++kk) {
            int r0 = kk * 4 + 2 * half;          // hw row for this lane's k pair
            v2f a, bf;
            a.x  = Qs[r0 + 0][q0 + m];           // A[m=q, k=hw]
            a.y  = Qs[r0 + 1][q0 + m];
            bf.x = Ks[r0 + 0][k0 + m];           // B[k=hw, n=k]
            bf.y = Ks[r0 + 1][k0 + m];
            acc = __builtin_amdgcn_wmma_f32_16x16x4_f32(
                false, a, false, bf, (short)0, acc, false, false);
        }
        __syncthreads();
    }

    // D layout: N = lane&15, M = vgpr + 8*half ; split-K reduce via L2 atomics
    float* S = scores + (size_t)b * CH * CH;
    #pragma unroll
    for (int g = 0; g < 8; ++g) {
        int M = g + 8 * half;
        atomicAdd(&S[(size_t)(q0 + M) * CH + (k0 + m)], acc[g]);
    }
}

// ---------------- phase 2: softmax over k, in place -------------------------
__global__ void softmax_kernel(float* __restrict__ scores) {
    const int b = blockIdx.x;     // 16
    const int q = threadIdx.x;    // 64
    float* row = scores + ((size_t)b * CH + q) * CH;
    float mx = row[0];
    #pragma unroll
    for (int k = 1; k < CH; ++k) mx = fmaxf(mx, row[k]);
    float s = 0.0f;
    #pragma unroll
    for (int k = 0; k < CH; ++k) {
        float e = __expf(row[k] - mx);
        row[k] = e;
        s += e;
    }
    float inv = 1.0f / s;
    #pragma unroll
    for (int k = 0; k < CH; ++k) row[k] *= inv;
}

// ---------------- phase 3: out[b,hw,q] = sum_k attn[b,q,k] * V[b,hw,k] ------
constexpr int P3_WGS   = 32;              // workgroups per batch
constexpr int P3_ROWS  = HW / P3_WGS;     // 512 hw rows per WG
constexpr int P3_CHUNK = 64;              // hw rows staged per iter

__global__ __launch_bounds__(512) void out_kernel(const float* __restrict__ V,
                                                  const float* __restrict__ attn,
                                                  float* __restrict__ out) {
    __shared__ float Vs[P3_CHUNK][LDSP];
    __shared__ float As[CH][LDSP];        // attn[q][k] for this batch

    const int b     = blockIdx.x / P3_WGS;
    const int split = blockIdx.x % P3_WGS;
    const int tid   = threadIdx.x;
    const int wave  = tid >> 5;
    const int lane  = tid & 31;
    const int m     = lane & 15;
    const int half  = lane >> 4;
    const int th    = wave >> 2;          // hw tile 0..3  (16 rows each)
    const int tq    = wave & 3;           // q  tile 0..3  (16 cols each)

    const float* Vb = V    + (size_t)b * HW * CH;
    const float* Ab = attn + (size_t)b * CH * CH;
    float*       Ob = out  + (size_t)b * HW * CH;

    // stage attn (64x64 = 1024 float4) once; attn is hot data -> regular loads
    #pragma unroll
    for (int i = 0; i < 2; ++i) {
        int t = tid + i * 512;
        int r = t >> 4;
        int c = (t & 15) * 4;
        *(f4*)&As[r][c] = *(const f4*)(Ab + (size_t)r * CH + c);
    }
    __syncthreads();

    const int hw_base = split * P3_ROWS;
    for (int c0 = 0; c0 < P3_ROWS; c0 += P3_CHUNK) {
        // stage 64x64 V chunk: 1024 float4, 2 per thread, nontemporal (streamed)
        #pragma unroll
        for (int i = 0; i < 2; ++i) {
            int t = tid + i * 512;
            int r = t >> 4;
            int c = (t & 15) * 4;
            f4 vv = __builtin_nontemporal_load(
                (const f4*)(Vb + (size_t)(hw_base + c0 + r) * CH + c));
            *(f4*)&Vs[r][c] = vv;
        }
        __syncthreads();

        v8f acc = {};
        #pragma unroll
        for (int kk = 0; kk < CH / 4; ++kk) {
            int kbase = kk * 4 + 2 * half;        // c index for this lane's pair
            v2f a, bf;
            a.x  = Vs[th * 16 + m][kbase + 0];    // A[m=hw, k=c] (stride-68 -> no conflicts)
            a.y  = Vs[th * 16 + m][kbase + 1];
            bf.x = As[tq * 16 + m][kbase + 0];    // B[k=c, n=q] = attn[q][c]
            bf.y = As[tq * 16 + m][kbase + 1];
            acc = __builtin_amdgcn_wmma_f32_16x16x4_f32(
                false, a, false, bf, (short)0, acc, false, false);
        }

        // coalesced nontemporal stores: N=q across lanes, M=hw across vgprs
        #pragma unroll
        for (int g = 0; g < 8; ++g) {
            int M = g + 8 * half;
            __builtin_nontemporal_store(
                acc[g],
                Ob + (size_t)(hw_base + c0 + th * 16 + M) * CH + tq * 16 + m);
        }
        __syncthreads();
    }
}

// ---------------- launch ----------------------------------------------------
extern "C" void kernel_launch(void* const* d_in, const int* in_sizes, int n_in,
                              void* d_out, int out_size, void* d_ws, size_t ws_size,
                              hipStream_t stream) {
    const float* Q = (const float*)d_in[0];
    const float* K = (const float*)d_in[1];
    const float* V = (const float*)d_in[2];
    float* out     = (float*)d_out;
    float* scores  = (float*)d_ws;        // 16*64*64 floats = 256 KB

    zero_scores_kernel<<<256, 256, 0, stream>>>(scores);
    scores_kernel<<<16 * P1_SPLITS, 512, 0, stream>>>(Q, K, scores);
    softmax_kernel<<<16, 64, 0, stream>>>(scores);
    out_kernel<<<16 * P3_WGS, 512, 0, stream>>>(V, scores, out);
}